// SSDMultiboxLoss_51290499449170
// MI455X (gfx1250) — compile-verified
//
#include <hip/hip_runtime.h>
#include <hip/hip_bf16.h>
#include <math.h>

#define BB 64
#define CC 9
#define NA 65536

typedef float v2f __attribute__((ext_vector_type(2)));
typedef float v8f __attribute__((ext_vector_type(8)));

// ---------------------------------------------------------------------------
// Wave32 sum reduction via V_WMMA_F32_16X16X4_F32 (codegen-confirmed round 1).
// A is 16x4 striped: lanes 0-15 hold {K0,K1}, lanes 16-31 hold {K2,K3}.
// Lane partial goes in the first A VGPR, second zeroed; B = ones(4x16).
// D[m][n] = p_m + p_{m+16}; column-sum of D = wave total, recovered from
// lanes 0 and 16. Requires all 32 lanes of the executing wave active.
// ---------------------------------------------------------------------------
__device__ __forceinline__ float wave_sum_wmma(float v) {
#if __has_builtin(__builtin_amdgcn_wmma_f32_16x16x4_f32)
  v2f a;  a[0] = v;    a[1] = 0.0f;
  v2f bm; bm[0] = 1.0f; bm[1] = 1.0f;
  v8f c = {};
  c = __builtin_amdgcn_wmma_f32_16x16x4_f32(
      /*neg_a=*/false, a, /*neg_b=*/false, bm,
      /*c_mod=*/(short)0, c, /*reuse_a=*/false, /*reuse_b=*/false);
  float s = c[0] + c[1] + c[2] + c[3] + c[4] + c[5] + c[6] + c[7];
  return __shfl(s, 0, 32) + __shfl(s, 16, 32);
#else
  #pragma unroll
  for (int off = 16; off > 0; off >>= 1) v += __shfl_down(v, off, 32);
  return __shfl(v, 0, 32);
#endif
}

// Block (256 threads) sum + one global f32 atomic. `red` is 256-float LDS.
__device__ __forceinline__ void block_sum_atomic(float v, float* target,
                                                 float* red) {
  const int tid = threadIdx.x;
  red[tid] = v;
  __syncthreads();
  if (tid < 32) {
    float s = 0.0f;
    #pragma unroll
    for (int i = 0; i < 8; ++i) s += red[tid * 8 + i];
    s = wave_sum_wmma(s);
    if (tid == 0) atomicAdd(target, s);
  }
  __syncthreads();
}

// Monotone float->uint key (order-preserving).
__device__ __forceinline__ unsigned fkey(float f) {
  unsigned u = __float_as_uint(f);
  return (u & 0x80000000u) ? ~u : (u | 0x80000000u);
}

// ---------------------------------------------------------------------------
// Pass A: single heavy-bandwidth pass, 4 anchors per thread, b128 traffic.
//  - log-softmax over C=9 -> ce, bg_loss (keys + ce cached to workspace)
//  - LDS-privatized 1024-bin histogram of negative bg keys, merged per batch
//  - smooth-L1 regression on positives, WMMA block reduction + 1 atomic
//  - positive count per batch, WMMA block reduction + 1 atomic
// Block = 256 threads = 1024 anchors; each block lies in one batch row.
// ---------------------------------------------------------------------------
__global__ void __launch_bounds__(256)
ssd_pass_a(const float* __restrict__ bbox_delta,
           const float* __restrict__ confs,
           const float* __restrict__ gt_bbox,
           const int*   __restrict__ gt_labels,
           const float* __restrict__ anchors,
           unsigned* __restrict__ keys,
           float*    __restrict__ cearr,
           unsigned* __restrict__ hist,
           float*    __restrict__ numposf,
           float*    __restrict__ regloss) {
  __shared__ unsigned lhist[1024];
  __shared__ float    red[256];
  const int tid = threadIdx.x;
  for (int i = tid; i < 1024; i += 256) lhist[i] = 0u;
  __syncthreads();

  const int idx4 = blockIdx.x * 1024 + tid * 4;  // first of 4 anchors
  const int b    = idx4 >> 16;                   // NA == 1<<16
  const int a    = idx4 & (NA - 1);

  // ---- wide loads ---------------------------------------------------------
  const float* cb = confs + (size_t)b * CC * NA + a;
  float4 xc[CC];
  #pragma unroll
  for (int c = 0; c < CC; ++c) xc[c] = *(const float4*)(cb + (size_t)c * NA);
  const float* xf = (const float*)&xc[0];                    // xf[c*4+j]

  const int4 lab4 = *(const int4*)(gt_labels + (size_t)b * NA + a);
  const int* labp = (const int*)&lab4;

  float4 an[4];
  #pragma unroll
  for (int k = 0; k < 4; ++k)
    an[k] = *(const float4*)(anchors + (size_t)k * NA + a);
  const float* anf = (const float*)&an[0];                   // anf[k*4+j]

  float4 dl[4];
  #pragma unroll
  for (int k = 0; k < 4; ++k)
    dl[k] = *(const float4*)(bbox_delta + (size_t)b * 4 * NA +
                             (size_t)k * NA + a);
  const float* dlf = (const float*)&dl[0];                   // dlf[k*4+j]

  float4 gt4[4];
  #pragma unroll
  for (int j = 0; j < 4; ++j)
    gt4[j] = *(const float4*)(gt_bbox + ((size_t)b * NA + a + j) * 4);

  // ---- per-anchor compute -------------------------------------------------
  unsigned kk[4];
  float    cee[4];
  float posn = 0.0f;
  float reg  = 0.0f;
  #pragma unroll
  for (int j = 0; j < 4; ++j) {
    float m = xf[0 * 4 + j];
    #pragma unroll
    for (int c = 1; c < CC; ++c) m = fmaxf(m, xf[c * 4 + j]);
    float se = 0.0f;
    #pragma unroll
    for (int c = 0; c < CC; ++c) se += __expf(xf[c * 4 + j] - m);
    const float lse = m + __logf(se);

    const int  label = labp[j];
    const bool pos   = label > 0;
    const float xl   = cb[(size_t)label * NA + j];  // cached reload, no scratch
    cee[j] = lse - xl;                              // -log_softmax[label]
    const float bg = lse - xf[0 * 4 + j];           // -log_softmax[0]

    unsigned key = 0u;
    if (!pos) {
      key = fkey(bg);
      atomicAdd(&lhist[key >> 22], 1u);
    } else {
      posn += 1.0f;
      const float ax = anf[0 * 4 + j], ay = anf[1 * 4 + j];
      const float aw = anf[2 * 4 + j], ah = anf[3 * 4 + j];
      const float gl0 = 10.0f * (gt4[j].x - ax) / aw;  // SCALE_XY = 10
      const float gl1 = 10.0f * (gt4[j].y - ay) / ah;
      const float gl2 = 5.0f * __logf(gt4[j].z / aw);  // SCALE_WH = 5
      const float gl3 = 5.0f * __logf(gt4[j].w / ah);
      const float d0 = dlf[0 * 4 + j] - gl0;
      const float d1 = dlf[1 * 4 + j] - gl1;
      const float d2 = dlf[2 * 4 + j] - gl2;
      const float d3 = dlf[3 * 4 + j] - gl3;
      #define SL1(d) (fabsf(d) < 1.0f ? 0.5f * (d) * (d) : fabsf(d) - 0.5f)
      reg += SL1(d0) + SL1(d1) + SL1(d2) + SL1(d3);
      #undef SL1
    }
    kk[j] = key;
  }

  // ---- wide stores of cached per-anchor results ---------------------------
  uint4 kv;  kv.x = kk[0];  kv.y = kk[1];  kv.z = kk[2];  kv.w = kk[3];
  float4 cv; cv.x = cee[0]; cv.y = cee[1]; cv.z = cee[2]; cv.w = cee[3];
  *(uint4*)(keys + idx4)   = kv;
  *(float4*)(cearr + idx4) = cv;

  // ---- merge LDS histogram into per-batch global histogram ---------------
  __syncthreads();
  for (int i = tid; i < 1024; i += 256) {
    const unsigned h = lhist[i];
    if (h) atomicAdd(&hist[(size_t)b * 1024 + i], h);
  }

  block_sum_atomic(reg, regloss, red);
  block_sum_atomic(posn, &numposf[b], red);
}

// ---------------------------------------------------------------------------
// Pass C: per-batch threshold select (64 threads, trivial).
// ---------------------------------------------------------------------------
__global__ void ssd_pass_c(const unsigned* __restrict__ hist,
                           const float*    __restrict__ numposf,
                           unsigned*       __restrict__ thresh) {
  const int b = threadIdx.x;
  if (b >= BB) return;
  const unsigned K = 3u * (unsigned)(numposf[b] + 0.5f);
  unsigned t;
  if (K == 0u) {
    t = 0xFFFFFFFFu;   // no negatives mined
  } else {
    unsigned cum = 0u;
    int bin = 1023;
    for (; bin > 0; --bin) {
      cum += hist[(size_t)b * 1024 + bin];
      if (cum >= K) break;
    }
    t = (unsigned)bin << 22;
  }
  thresh[b] = t;
}

// ---------------------------------------------------------------------------
// Pass D: classification sum over pos | (key >= thresh).
// Streams keys/ce/labels through LDS with GLOBAL_LOAD_ASYNC_TO_LDS_B128
// (ASYNCcnt-tracked, no VGPR staging). Each lane consumes only the 16B
// slices its own async ops wrote, so a per-wave s_wait_asynccnt suffices.
// LDS byte address = low 32 bits of the generic pointer (ISA aperture rule).
// ---------------------------------------------------------------------------
__global__ void __launch_bounds__(256)
ssd_pass_d(const unsigned* __restrict__ keys,
           const float*    __restrict__ cearr,
           const int*      __restrict__ gt_labels,
           const unsigned* __restrict__ thresh,
           float*          __restrict__ clsloss) {
  __shared__ unsigned skeys[1024];
  __shared__ float    sce[1024];
  __shared__ int      slab[1024];
  __shared__ float    red[256];
  const int tid  = threadIdx.x;
  const int idx4 = blockIdx.x * 1024 + tid * 4;
  const int b    = idx4 >> 16;

  {
    const unsigned lk = (unsigned)(uintptr_t)(void*)&skeys[tid * 4];
    const unsigned lc = (unsigned)(uintptr_t)(void*)&sce[tid * 4];
    const unsigned ll = (unsigned)(uintptr_t)(void*)&slab[tid * 4];
    const unsigned long long gk = (unsigned long long)(uintptr_t)(keys + idx4);
    const unsigned long long gc = (unsigned long long)(uintptr_t)(cearr + idx4);
    const unsigned long long gl =
        (unsigned long long)(uintptr_t)(gt_labels + idx4);
    asm volatile(
        "global_load_async_to_lds_b128 %0, %3, off\n\t"
        "global_load_async_to_lds_b128 %1, %4, off\n\t"
        "global_load_async_to_lds_b128 %2, %5, off\n\t"
        "s_wait_asynccnt 0x0"
        :
        : "v"(lk), "v"(lc), "v"(ll), "v"(gk), "v"(gc), "v"(gl)
        : "memory");
  }

  const unsigned t = thresh[b];
  float part = 0.0f;
  #pragma unroll
  for (int j = 0; j < 4; ++j) {
    const unsigned key = skeys[tid * 4 + j];
    const float    ce  = sce[tid * 4 + j];
    const bool     pos = slab[tid * 4 + j] > 0;
    if (pos || key >= t) part += ce;
  }
  block_sum_atomic(part, clsloss, red);
}

// Finalize: total = (reg + cls) / n_pos. One thread.
__global__ void ssd_finalize(const float* __restrict__ numposf,
                             const float* __restrict__ regloss,
                             const float* __restrict__ clsloss,
                             float* __restrict__ out) {
  float np = 0.0f;
  for (int i = 0; i < BB; ++i) np += numposf[i];
  out[0] = (regloss[0] + clsloss[0]) / np;
}

__global__ void ssd_zero(unsigned* __restrict__ p, int n) {
  const int i = blockIdx.x * 256 + threadIdx.x;
  if (i < n) p[i] = 0u;
}

extern "C" void kernel_launch(void* const* d_in, const int* in_sizes, int n_in,
                              void* d_out, int out_size, void* d_ws,
                              size_t ws_size, hipStream_t stream) {
  const float* bbox_delta = (const float*)d_in[0];   // [B,4,NA]
  const float* confs      = (const float*)d_in[1];   // [B,C,NA]
  const float* gt_bbox    = (const float*)d_in[2];   // [B,NA,4]
  const int*   gt_labels  = (const int*)  d_in[3];   // [B,NA]
  const float* anchors    = (const float*)d_in[4];   // [1,4,NA]
  float* out = (float*)d_out;

  // Workspace layout (bytes):
  //   [0)          keys   : uint[B*NA]    16 MB
  //   [B*NA*4)     cearr  : float[B*NA]   16 MB
  //   [2*B*NA*4)   hist   : uint[B*1024]  256 KB } zeroed
  //   ...          numposf: float[B]             } zeroed
  //   ...          thresh : uint[B]              } zeroed
  //   ...          regloss, clsloss : float      } zeroed
  char* ws = (char*)d_ws;
  unsigned* keys    = (unsigned*)ws;
  float*    cearr   = (float*)(ws + (size_t)BB * NA * 4);
  unsigned* hist    = (unsigned*)(ws + (size_t)2 * BB * NA * 4);
  float*    numposf = (float*)((char*)hist + (size_t)BB * 1024 * 4);
  unsigned* thresh  = (unsigned*)((char*)numposf + BB * 4);
  float*    regloss = (float*)((char*)thresh + BB * 4);
  float*    clsloss = regloss + 1;

  const int zero_words = BB * 1024 + BB + BB + 2;  // hist..clsloss contiguous
  ssd_zero<<<(zero_words + 255) / 256, 256, 0, stream>>>(hist, zero_words);

  const int nblocks = (BB * NA) / 1024;  // 4096 (4 anchors/thread)
  ssd_pass_a<<<nblocks, 256, 0, stream>>>(bbox_delta, confs, gt_bbox,
                                          gt_labels, anchors, keys, cearr,
                                          hist, numposf, regloss);
  ssd_pass_c<<<1, 64, 0, stream>>>(hist, numposf, thresh);
  ssd_pass_d<<<nblocks, 256, 0, stream>>>(keys, cearr, gt_labels, thresh,
                                          clsloss);
  ssd_finalize<<<1, 1, 0, stream>>>(numposf, regloss, clsloss, out);
}